// ClusterLoss_15427522527361
// MI455X (gfx1250) — compile-verified
//
#include <hip/hip_runtime.h>
#include <math.h>

#define NCLS   64
#define DIM    192
#define EPSV   1e-6f

// Workspace layout (floats):
//   [0, NCLS*DIM)        per-class sums
//   [NCLS*DIM, +NCLS)    per-class counts
//   [NCLS*DIM+NCLS]      intra-class scalar accumulator
#define WS_SUMS  0
#define WS_CNT   (NCLS * DIM)
#define WS_INTRA (NCLS * DIM + NCLS)
#define WS_TOTAL (NCLS * DIM + NCLS + 1)

typedef __attribute__((ext_vector_type(2))) float v2f;
typedef __attribute__((ext_vector_type(8))) float v8f;

__global__ void zero_ws_kernel(float* __restrict__ ws) {
    int i = blockIdx.x * blockDim.x + threadIdx.x;
    if (i < WS_TOTAL) ws[i] = 0.0f;
}

// Pass 1: per-class sums/counts. Each block accumulates into a private 48KB
// LDS tile with ds_add_f32 atomics, then flushes once with global f32 atomics.
__global__ void class_sums_kernel(const float* __restrict__ x,
                                  const long long* __restrict__ lab,
                                  float* __restrict__ ws, int N) {
    __shared__ float s_sum[NCLS * DIM];
    __shared__ float s_cnt[NCLS];
    for (int i = threadIdx.x; i < NCLS * DIM; i += blockDim.x) s_sum[i] = 0.0f;
    if (threadIdx.x < NCLS) s_cnt[threadIdx.x] = 0.0f;
    __syncthreads();

    const int lane        = threadIdx.x & 31;
    const int wave        = threadIdx.x >> 5;
    const int wavesPerBlk = blockDim.x >> 5;
    const int gwave       = blockIdx.x * wavesPerBlk + wave;
    const int nwaves      = gridDim.x * wavesPerBlk;

    for (int row = gwave; row < N; row += nwaves) {
        const int c = (int)lab[row];
        const float* xr = x + (size_t)row * DIM;
        float* dst = s_sum + c * DIM;
#pragma unroll
        for (int k = 0; k < DIM / 32; ++k) {
            // 32 lanes touch 32 consecutive dwords: coalesced load, conflict-free LDS
            atomicAdd(&dst[lane + k * 32], xr[lane + k * 32]);
        }
        if (lane == 0) atomicAdd(&s_cnt[c], 1.0f);
    }
    __syncthreads();
    for (int i = threadIdx.x; i < NCLS * DIM; i += blockDim.x)
        atomicAdd(&ws[WS_SUMS + i], s_sum[i]);
    if (threadIdx.x < NCLS)
        atomicAdd(&ws[WS_CNT + threadIdx.x], s_cnt[threadIdx.x]);
}

// Pass 2: intra-class term. Means cached in LDS; one wave32 per row,
// lane-strided coalesced loads, shfl_xor tree reduction, sqrt per row,
// one global atomicAdd per wave at the end.
__global__ void intra_kernel(const float* __restrict__ x,
                             const long long* __restrict__ lab,
                             const float* __restrict__ ws,
                             float* __restrict__ ws_acc, int N) {
    __shared__ float s_mean[NCLS * DIM];
    for (int i = threadIdx.x; i < NCLS * DIM; i += blockDim.x)
        s_mean[i] = ws[WS_SUMS + i] / ws[WS_CNT + i / DIM];
    __syncthreads();

    const int lane        = threadIdx.x & 31;
    const int wave        = threadIdx.x >> 5;
    const int wavesPerBlk = blockDim.x >> 5;
    const int gwave       = blockIdx.x * wavesPerBlk + wave;
    const int nwaves      = gridDim.x * wavesPerBlk;

    float local = 0.0f;
    for (int row = gwave; row < N; row += nwaves) {
        const int c = (int)lab[row];
        const float* xr = x + (size_t)row * DIM;
        const float* mr = s_mean + c * DIM;
        float acc = 0.0f;
#pragma unroll
        for (int k = 0; k < DIM / 32; ++k) {
            float d = xr[lane + k * 32] - mr[lane + k * 32] + EPSV;
            acc += d * d;
        }
#pragma unroll
        for (int off = 16; off > 0; off >>= 1) acc += __shfl_xor(acc, off, 32);
        if (lane == 0) local += sqrtf(acc);
    }
    if (lane == 0) atomicAdd(&ws_acc[WS_INTRA], local);
}

// Pass 3: inter-class term via WMMA Gram matrix G = M * M^T (64x192x64, f32),
// then dist_ij = sqrt(G_ii + G_jj - 2 G_ij + 2*eps*(s_i - s_j) + D*eps^2).
// Single block of 16 waves; each wave owns one 16x16 tile of G and drives
// V_WMMA_F32_16X16X4_F32 over 48 K-steps.
#define MPITCH 196  // 192 % 64 == 0 would put all 16 A-rows in one LDS bank

__global__ void inter_final_kernel(const float* __restrict__ ws,
                                   float* __restrict__ out) {
    __shared__ float s_m[NCLS * MPITCH];
    __shared__ float s_n2[NCLS];   // ||m_c||^2  (diag of G)
    __shared__ float s_rs[NCLS];   // sum_d m_c[d]
    __shared__ float s_red[16];

    for (int i = threadIdx.x; i < NCLS * DIM; i += blockDim.x) {
        int c = i / DIM, d = i % DIM;
        s_m[c * MPITCH + d] = ws[WS_SUMS + i] / ws[WS_CNT + c];
    }
    __syncthreads();
    if (threadIdx.x < NCLS) {
        const float* mr = s_m + threadIdx.x * MPITCH;
        float n2 = 0.0f, rs = 0.0f;
        for (int d = 0; d < DIM; ++d) { n2 += mr[d] * mr[d]; rs += mr[d]; }
        s_n2[threadIdx.x] = n2;
        s_rs[threadIdx.x] = rs;
    }
    __syncthreads();

    const int lane = threadIdx.x & 31;
    const int wave = threadIdx.x >> 5;          // 16 waves -> 4x4 tile grid
    const int trow = (wave >> 2) * 16;
    const int tcol = (wave & 3) * 16;
    const int half = lane >> 4;                 // K sub-pair select per ISA layout
    const int l15  = lane & 15;

    // A: 16x4 f32 of rows [trow,trow+16); B: 4x16 f32 = columns of M^T, i.e.
    // rows [tcol,tcol+16) of M — same gather pattern as A.
    v8f acc = {};
    for (int kb = 0; kb < DIM; kb += 4) {
        const int k0 = kb + half * 2;
        v2f a, b;
        a.x = s_m[(trow + l15) * MPITCH + k0];
        a.y = s_m[(trow + l15) * MPITCH + k0 + 1];
        b.x = s_m[(tcol + l15) * MPITCH + k0];
        b.y = s_m[(tcol + l15) * MPITCH + k0 + 1];
        acc = __builtin_amdgcn_wmma_f32_16x16x4_f32(
            /*neg_a=*/false, a, /*neg_b=*/false, b,
            /*c_mod=*/(short)0, acc, /*reuse_a=*/false, /*reuse_b=*/false);
    }

    // C/D layout: VGPR v, lanes 0-15 -> M=v, lanes 16-31 -> M=v+8; N = lane&15.
    float sum = 0.0f;
#pragma unroll
    for (int v = 0; v < 8; ++v) {
        const int i = trow + half * 8 + v;
        const int j = tcol + l15;
        if (i != j) {
            float d2 = s_n2[i] + s_n2[j] - 2.0f * acc[v]
                     + 2.0f * EPSV * (s_rs[i] - s_rs[j])
                     + (float)DIM * EPSV * EPSV;
            sum += sqrtf(fmaxf(d2, 0.0f));
        }
    }
#pragma unroll
    for (int off = 16; off > 0; off >>= 1) sum += __shfl_xor(sum, off, 32);
    if (lane == 0) s_red[wave] = sum;
    __syncthreads();
    if (threadIdx.x == 0) {
        float inter = 0.0f;
        for (int w = 0; w < 16; ++w) inter += s_red[w];
        out[0] = ws[WS_INTRA] - inter;
    }
}

extern "C" void kernel_launch(void* const* d_in, const int* in_sizes, int n_in,
                              void* d_out, int out_size, void* d_ws, size_t ws_size,
                              hipStream_t stream) {
    const float*     x   = (const float*)d_in[0];
    const long long* lab = (const long long*)d_in[1];  // reference dtype: int64
    float* ws  = (float*)d_ws;
    float* out = (float*)d_out;
    const int N = in_sizes[1];  // number of labels == number of rows

    zero_ws_kernel<<<(WS_TOTAL + 255) / 256, 256, 0, stream>>>(ws);
    class_sums_kernel<<<1024, 256, 0, stream>>>(x, lab, ws, N);
    intra_kernel<<<1024, 256, 0, stream>>>(x, lab, ws, ws, N);
    inter_final_kernel<<<1, 512, 0, stream>>>(ws, out);
}